// PerfectTreeTraversalTreeImpl_82884278878828
// MI455X (gfx1250) — compile-verified
//
#include <hip/hip_runtime.h>
#include <stdint.h>

#define NUM_TREES   512
#define N_FEATURES  256
#define BATCH       65536
#define TILE_S      32            // samples per block
#define ROW_STRIDE  257           // 256 + 1 dword pad: root gathers conflict-free
#define BLOCK       128           // 4 waves (wave32)
#define WAVES       (BLOCK / 32)
#define TREES_PER_WAVE (NUM_TREES / WAVES)   // 128
#define CHUNK       32            // trees buffered per transpose pass
#define TBUF_STRIDE 33            // padded: conflict-free transpose

typedef __attribute__((address_space(1))) int global_int_t;
typedef __attribute__((address_space(3))) int lds_int_t;

// ---- CDNA5 async global->LDS copy (ASYNCcnt-tracked) -----------------------
__device__ __forceinline__ void async_copy_dword(const float* g, float* l) {
#if __has_builtin(__builtin_amdgcn_global_load_async_to_lds_b32)
  __builtin_amdgcn_global_load_async_to_lds_b32(
      (global_int_t*)(uintptr_t)g,
      (lds_int_t*)l,
      /*imm offset*/ 0, /*cpol*/ 0);
#else
  asm volatile("global_load_async_to_lds_b32 %0, %1, off"
               :
               : "v"((lds_int_t*)l),
                 "v"((global_int_t*)(uintptr_t)g)
               : "memory");
#endif
}

__device__ __forceinline__ void wait_async_zero() {
#if __has_builtin(__builtin_amdgcn_s_wait_asynccnt)
  __builtin_amdgcn_s_wait_asynccnt(0);
#else
  asm volatile("s_wait_asynccnt 0" ::: "memory");
#endif
}
// ---------------------------------------------------------------------------

__global__ __launch_bounds__(BLOCK) void tree_traverse_kernel(
    const float* __restrict__ x,            // (BATCH, 256)
    const int*   __restrict__ root_nodes,   // (512,)
    const float* __restrict__ root_biases,  // (512,)
    const int*   __restrict__ f1, const float* __restrict__ t1,
    const int*   __restrict__ f2, const float* __restrict__ t2,
    const int*   __restrict__ f3, const float* __restrict__ t3,
    const int*   __restrict__ f4, const float* __restrict__ t4,
    const int*   __restrict__ f5, const float* __restrict__ t5,
    const float* __restrict__ leaves,       // (512*64,)
    float*       __restrict__ out)          // (BATCH, 512)
{
  __shared__ float xtile[TILE_S * ROW_STRIDE];          // 32.9 KB
  __shared__ float tbuf[WAVES][TILE_S * TBUF_STRIDE];   // 16.9 KB

  const int tid  = threadIdx.x;
  const int lane = tid & 31;
  const int w    = tid >> 5;
  const int b0   = blockIdx.x * TILE_S;

  // ---- Stage 32 sample rows (32 KB) into LDS via async DMA-style copies ----
  // 8192 dwords / 128 threads = 64 per thread; consecutive tids hit
  // consecutive dwords -> 128B-coalesced global reads per wave instruction.
  {
    const float* xbase = x + (size_t)b0 * N_FEATURES;
    #pragma unroll 4
    for (int k = 0; k < (TILE_S * N_FEATURES) / BLOCK; ++k) {
      const int d   = k * BLOCK + tid;
      const int row = d >> 8;        // /256
      const int col = d & 255;
      async_copy_dword(xbase + row * N_FEATURES + col,
                       &xtile[row * ROW_STRIDE + col]);
    }
  }
  wait_async_zero();     // drain this wave's ASYNCcnt
  __syncthreads();       // make every wave's staged rows visible

  const int*   feats[5] = {f1, f2, f3, f4, f5};
  const float* thrs[5]  = {t1, t2, t3, t4, t5};

  const float* xrow = &xtile[lane * ROW_STRIDE];   // lane = sample
  float*       tb   = &tbuf[w][0];                 // per-wave, no barrier needed
  const int wtree0  = __builtin_amdgcn_readfirstlane(w * TREES_PER_WAVE);

  for (int c = 0; c < TREES_PER_WAVE / CHUNK; ++c) {
    const int tbase = wtree0 + c * CHUNK;

    // Traverse 32 trees for this lane's sample; results into padded LDS tile.
    for (int j = 0; j < CHUNK; ++j) {
      const int t = tbase + j;                     // wave-uniform -> scalar loads
      const float fv0 = xrow[root_nodes[t]];       // bank = (lane+feat)%64: conflict-free
      int prev = 2 * t + (fv0 >= root_biases[t] ? 1 : 0);
      #pragma unroll
      for (int l = 0; l < 5; ++l) {
        const int   f  = feats[l][prev];           // <=256B window per wave: L0-hot
        const float th = thrs[l][prev];
        const float v  = xrow[f];                  // LDS gather
        prev = 2 * prev + (v >= th ? 1 : 0);
      }
      tb[lane * TBUF_STRIDE + j] = leaves[prev];   // padded: conflict-free write
    }

    // Transpose through LDS -> fully coalesced 128B stores along the tree axis.
    const int tcol = tbase + lane;
    #pragma unroll 4
    for (int s = 0; s < TILE_S; ++s) {
      out[(size_t)(b0 + s) * NUM_TREES + tcol] = tb[s * TBUF_STRIDE + lane];
    }
  }
}

extern "C" void kernel_launch(void* const* d_in, const int* in_sizes, int n_in,
                              void* d_out, int out_size, void* d_ws, size_t ws_size,
                              hipStream_t stream) {
  (void)in_sizes; (void)n_in; (void)out_size; (void)d_ws; (void)ws_size;
  const float* x           = (const float*)d_in[0];
  const int*   root_nodes  = (const int*)  d_in[1];
  const float* root_biases = (const float*)d_in[2];
  const int*   f1 = (const int*)d_in[3];  const float* t1 = (const float*)d_in[4];
  const int*   f2 = (const int*)d_in[5];  const float* t2 = (const float*)d_in[6];
  const int*   f3 = (const int*)d_in[7];  const float* t3 = (const float*)d_in[8];
  const int*   f4 = (const int*)d_in[9];  const float* t4 = (const float*)d_in[10];
  const int*   f5 = (const int*)d_in[11]; const float* t5 = (const float*)d_in[12];
  const float* leaves      = (const float*)d_in[13];
  float* out = (float*)d_out;

  dim3 grid(BATCH / TILE_S);   // 2048 blocks
  dim3 block(BLOCK);           // 128 threads = 4 waves
  tree_traverse_kernel<<<grid, block, 0, stream>>>(
      x, root_nodes, root_biases,
      f1, t1, f2, t2, f3, t3, f4, t4, f5, t5,
      leaves, out);
}